// LIF_residue_45621142618289
// MI455X (gfx1250) — compile-verified
//
#include <hip/hip_runtime.h>
#include <cstdint>

// LIF-with-residue sequential scan over 128 timesteps, elementwise over 131072
// lanes. Bandwidth-bound (128 MiB @ 23.3 TB/s ~ 5.6 us). CDNA5 path: deep
// async global->LDS pipeline (ASYNCcnt) so each wave keeps STAGES*512B in
// flight without VGPR cost. Each wave produces and consumes only its own LDS
// slice -> no barriers.

#define TPB     128          // threads per block (4 waves)
#define STAGES  16           // async pipeline depth (power of two)
#define STEPS   128          // steps * T = 32 * 4

#if defined(__has_builtin)
#  if __has_builtin(__builtin_amdgcn_global_load_async_to_lds_b128)
#    define HAVE_ASYNC_LDS_BUILTIN 1
#  endif
#  if __has_builtin(__builtin_amdgcn_s_wait_asynccnt)
#    define HAVE_WAIT_ASYNC_BUILTIN 1
#  endif
#  if __has_builtin(__builtin_amdgcn_s_wait_dscnt)
#    define HAVE_WAIT_DS_BUILTIN 1
#  endif
#endif

// 128-bit payload type the builtin expects (per hipcc diagnostic:
// "__attribute__((__vector_size__(4 * sizeof(int)))) int __device__ *").
typedef int v4i __attribute__((vector_size(16)));
typedef __attribute__((address_space(1))) v4i* glb_v4i_p;   // global
typedef __attribute__((address_space(3))) v4i* lds_v4i_p;   // LDS

// Issue one async 16B/lane global->LDS copy (tracked by ASYNCcnt).
__device__ __forceinline__ void async_copy_b128(const void* gsrc, void* ldst) {
#if defined(HAVE_ASYNC_LDS_BUILTIN)
  // (global v4i* src, lds v4i* dst, imm offset, imm cpol)
  __builtin_amdgcn_global_load_async_to_lds_b128(
      (glb_v4i_p)(uintptr_t)gsrc, (lds_v4i_p)(uintptr_t)ldst, 0, 0);
#else
  // Low 32 bits of the flat address of a __shared__ object == LDS byte offset.
  unsigned lds_off = (unsigned)(uintptr_t)ldst;
  asm volatile("global_load_async_to_lds_b128 %0, %1, off"
               :: "v"(lds_off), "v"(gsrc)
               : "memory");
#endif
}

#if defined(HAVE_WAIT_ASYNC_BUILTIN)
#  define WAIT_ASYNCCNT(n)                                     \
    do { asm volatile("" ::: "memory");                        \
         __builtin_amdgcn_s_wait_asynccnt(n);                  \
         asm volatile("" ::: "memory"); } while (0)
#else
#  define WAIT_ASYNCCNT(n) asm volatile("s_wait_asynccnt " #n ::: "memory")
#endif

#if defined(HAVE_WAIT_DS_BUILTIN)
#  define WAIT_DSCNT0()                                        \
    do { asm volatile("" ::: "memory");                        \
         __builtin_amdgcn_s_wait_dscnt(0);                     \
         asm volatile("" ::: "memory"); } while (0)
#else
#  define WAIT_DSCNT0() asm volatile("s_wait_dscnt 0" ::: "memory")
#endif

// One LIF step on one scalar. THRESH=TAU=GAMA=1, ALPHA=0.5.
// spike is exactly 0.0f or 1.0f; (mem >= 1.0f) == (mem - 1.0f >= 0).
__device__ __forceinline__ void lif_step(float xin, float& mem, float& res) {
  float m     = mem + xin;
  float spike = (m >= 1.0f) ? 1.0f : 0.0f;
  res = fmaf(0.5f, res, spike);       // residue = 0.5*residue + spike*THRESH
  mem = m * (1.0f - spike);           // reset on spike (matches reference exactly)
}

__device__ __forceinline__ void lif_step4(const float4& v, float4& mem, float4& res) {
  lif_step(v.x, mem.x, res.x);
  lif_step(v.y, mem.y, res.y);
  lif_step(v.z, mem.z, res.z);
  lif_step(v.w, mem.w, res.w);
}

__global__ __launch_bounds__(TPB)
void lif_residue_scan_kernel(const float4* __restrict__ x,
                             float4* __restrict__ out,
                             int nv4) {
  __shared__ float4 buf[STAGES * TPB];   // 16 stages * 128 lanes * 16B = 32 KB

  const int tid  = threadIdx.x;
  const int slot = blockIdx.x * TPB + tid;   // this thread's float4 column
  if (slot >= nv4) return;

  // ---- prologue: fill the async pipeline with timesteps 0..STAGES-1 ----
#pragma unroll
  for (int s = 0; s < STAGES; ++s) {
    async_copy_b128(&x[(size_t)s * nv4 + slot], &buf[s * TPB + tid]);
  }

  float4 mem = make_float4(0.f, 0.f, 0.f, 0.f);
  float4 res = make_float4(0.f, 0.f, 0.f, 0.f);

  // ---- steady state: consume oldest stage, refill with t+STAGES ----
  for (int t = 0; t < STEPS - STAGES; ++t) {
    WAIT_ASYNCCNT(15);                     // STAGES-1: oldest stage has landed
    const int st = t & (STAGES - 1);
    float4 v = buf[st * TPB + tid];        // ds_load_b128
    lif_step4(v, mem, res);
    out[(size_t)t * nv4 + slot] = res;     // global_store_b128
    WAIT_DSCNT0();                         // WAR: v is in VGPRs before refill
    async_copy_b128(&x[(size_t)(t + STAGES) * nv4 + slot], &buf[st * TPB + tid]);
  }

  // ---- drain: everything already in LDS ----
  WAIT_ASYNCCNT(0);
  for (int t = STEPS - STAGES; t < STEPS; ++t) {
    const int st = t & (STAGES - 1);
    float4 v = buf[st * TPB + tid];
    lif_step4(v, mem, res);
    out[(size_t)t * nv4 + slot] = res;
  }
}

extern "C" void kernel_launch(void* const* d_in, const int* in_sizes, int n_in,
                              void* d_out, int out_size, void* d_ws, size_t ws_size,
                              hipStream_t stream) {
  (void)n_in; (void)out_size; (void)d_ws; (void)ws_size;
  const float* x = (const float*)d_in[0];
  float* out     = (float*)d_out;

  const int total = in_sizes[0];        // steps * T*B * D = 16,777,216
  const int nelem = total / STEPS;      // elements per timestep = 131,072
  const int nv4   = nelem / 4;          // float4 columns      = 32,768
  const int blocks = (nv4 + TPB - 1) / TPB;   // 256 blocks

  lif_residue_scan_kernel<<<blocks, TPB, 0, stream>>>(
      (const float4*)x, (float4*)out, nv4);
}